// ChebGCN1_63024350101692
// MI455X (gfx1250) — compile-verified
//
#include <hip/hip_runtime.h>
#include <math.h>

// ---------------- problem constants (match reference) ----------------
#define N_NODES 20000
#define N_EDGES 320000
#define DIM     256
#define KCHEB   4
#define KDIM    (KCHEB*DIM)   // 1024
#define OUTF    10
#define NLAYERS 4
#define EPS_GN  1e-5f
#define STAT_ROWS 128
#define STAT_BLOCKS ((N_NODES + STAT_ROWS - 1) / STAT_ROWS)   // 157

typedef float v2f __attribute__((ext_vector_type(2)));
typedef float v8f __attribute__((ext_vector_type(8)));

// ---------------- small utility kernels ----------------
__global__ void k_zero_i32(int* __restrict__ p, int n) {
  int i = blockIdx.x * blockDim.x + threadIdx.x;
  if (i < n) p[i] = 0;
}

__global__ void k_copy_i32(int* __restrict__ dst, const int* __restrict__ src, int n) {
  int i = blockIdx.x * blockDim.x + threadIdx.x;
  if (i < n) dst[i] = src[i];
}

// integer degree histogram (order-independent -> deterministic)
__global__ void k_deg(const int* __restrict__ row, int* __restrict__ degi) {
  int e = blockIdx.x * blockDim.x + threadIdx.x;
  if (e < N_EDGES) atomicAdd(&degi[row[e]], 1);
}

// dis[i] = deg>0 ? deg^-1/2 : 0
__global__ void k_dis(const int* __restrict__ degi, float* __restrict__ dis) {
  int i = blockIdx.x * blockDim.x + threadIdx.x;
  if (i < N_NODES) {
    int d = degi[i];
    dis[i] = (d > 0) ? rsqrtf((float)d) : 0.0f;
  }
}

// w_edge[e] = -dis[row]*dis[col]
__global__ void k_wedge(const int* __restrict__ row, const int* __restrict__ col,
                        const float* __restrict__ dis, float* __restrict__ w) {
  int e = blockIdx.x * blockDim.x + threadIdx.x;
  if (e < N_EDGES) w[e] = -dis[row[e]] * dis[col[e]];
}

// single-block chunked exclusive scan of degi[N_NODES] -> rowptr[N_NODES+1]
__global__ void k_scan(const int* __restrict__ degi, int* __restrict__ rowptr) {
  __shared__ int sh[256];
  __shared__ int carry;
  int t = threadIdx.x;
  if (t == 0) carry = 0;
  __syncthreads();
  for (int base = 0; base < N_NODES; base += 256) {
    int i = base + t;
    int v = (i < N_NODES) ? degi[i] : 0;
    sh[t] = v;
    __syncthreads();
    for (int off = 1; off < 256; off <<= 1) {   // Hillis-Steele inclusive scan
      int add = (t >= off) ? sh[t - off] : 0;
      __syncthreads();
      sh[t] += add;
      __syncthreads();
    }
    if (i < N_NODES) rowptr[i] = carry + sh[t] - v;   // exclusive
    __syncthreads();
    if (t == 255) carry += sh[255];
    __syncthreads();
  }
  if (t == 0) rowptr[N_NODES] = carry;   // == N_EDGES
}

// scatter edges into CSR slots (int atomics pick slots; per-row sets identical each run)
__global__ void k_csr_fill(const int* __restrict__ row, const int* __restrict__ col,
                           const float* __restrict__ wed, int* __restrict__ cursor,
                           int* __restrict__ csr_col, float* __restrict__ csr_w) {
  int e = blockIdx.x * blockDim.x + threadIdx.x;
  if (e < N_EDGES) {
    int r = row[e];
    int p = atomicAdd(&cursor[r], 1);
    csr_col[p] = col[e];
    csr_w[p]   = wed[e];
  }
}

// dst[n*ldd + d] = src[n*lds + d]   (over N_NODES x DIM)
__global__ void k_copy(float* __restrict__ dst, int ldd,
                       const float* __restrict__ src, int lds) {
  int i = blockIdx.x * blockDim.x + threadIdx.x;
  if (i < N_NODES * DIM) {
    int n = i >> 8, d = i & 255;
    dst[(size_t)n * ldd + d] = src[(size_t)n * lds + d];
  }
}

// ---------------- CSR gather SpMM (no float atomics) ----------------
// dst[n,:] = alpha * sum_j w_j * src[col_j,:]  (+ beta * prev[n,:])
// One wave per node; lane owns 8 features as two float4 lanesets. All L2-resident.
#define FMA4(acc, w, v) { acc.x += (w)*(v).x; acc.y += (w)*(v).y; \
                          acc.z += (w)*(v).z; acc.w += (w)*(v).w; }
__global__ void k_prop(float* __restrict__ dst, const float* __restrict__ src,
                       const float* __restrict__ prev,
                       const int* __restrict__ rowptr, const int* __restrict__ csr_col,
                       const float* __restrict__ csr_w, float alpha, float beta) {
  int gid  = blockIdx.x * blockDim.x + threadIdx.x;
  int n    = gid >> 5;
  int lane = threadIdx.x & 31;
  if (n >= N_NODES) return;
  int p0 = rowptr[n], p1 = rowptr[n + 1];
  float4 a0 = {0, 0, 0, 0}, a1 = {0, 0, 0, 0};
  for (int p = p0; p < p1; ++p) {
    int c   = csr_col[p];
    float w = csr_w[p];
    const float4* s = (const float4*)(src + (size_t)c * KDIM);
    float4 v0 = s[lane];
    float4 v1 = s[lane + 32];
    FMA4(a0, w, v0);
    FMA4(a1, w, v1);
  }
  float4* d4 = (float4*)(dst + (size_t)n * KDIM);
  if (beta != 0.0f) {
    const float4* pb = (const float4*)(prev + (size_t)n * KDIM);
    float4 q0 = pb[lane], q1 = pb[lane + 32];
    a0.x = alpha * a0.x + beta * q0.x;  a0.y = alpha * a0.y + beta * q0.y;
    a0.z = alpha * a0.z + beta * q0.z;  a0.w = alpha * a0.w + beta * q0.w;
    a1.x = alpha * a1.x + beta * q1.x;  a1.y = alpha * a1.y + beta * q1.y;
    a1.z = alpha * a1.z + beta * q1.z;  a1.w = alpha * a1.w + beta * q1.w;
  } else {
    a0.x *= alpha; a0.y *= alpha; a0.z *= alpha; a0.w *= alpha;
    a1.x *= alpha; a1.y *= alpha; a1.z *= alpha; a1.w *= alpha;
  }
  d4[lane]      = a0;
  d4[lane + 32] = a1;
}

// ---------------- WMMA fp32 GEMM ----------------
// C[N x DIM] = A[N x KDIM] @ B[KDIM x DIM] + bias, fp32 V_WMMA_F32_16X16X4_F32.
// Wave computes a 16x64 strip: one A fragment feeds 4 independent accumulators
// (4 interleaved D->C chains hide WMMA latency). Block: 8 waves = 32 rows x 256 cols.
__global__ void k_gemm_wmma(const float* __restrict__ A,
                            const float* __restrict__ B,
                            const float* __restrict__ bias,
                            float* __restrict__ C) {
  int lane = threadIdx.x & 31;
  int wave = threadIdx.x >> 5;          // 0..7
  int half = lane >> 4;                 // 0 or 1
  int lr   = lane & 15;
  int tm   = blockIdx.x * 2 + (wave >> 2);   // row tile, 1250 total
  int ng   = wave & 3;                       // 64-col group
  int arow = tm * 16 + lr;
  int col0 = ng * 64 + lr;

  const float* ap = A + (size_t)arow * KDIM + 2 * half;  // a.x=K, a.y=K+1
  const float* bp = B + (size_t)(2 * half) * DIM + col0;

  v8f c0, c1, c2, c3;
  float bv0 = bias[col0], bv1 = bias[col0 + 16], bv2 = bias[col0 + 32], bv3 = bias[col0 + 48];
#pragma unroll
  for (int v = 0; v < 8; ++v) { c0[v] = bv0; c1[v] = bv1; c2[v] = bv2; c3[v] = bv3; }

  for (int k = 0; k < KDIM; k += 4) {
    float2 af = *(const float2*)(ap + k);
    v2f a;  a.x = af.x;  a.y = af.y;
    const float* brow = bp + (size_t)k * DIM;
    v2f b0, b1, b2, b3;
    b0.x = brow[0];   b0.y = brow[DIM];
    b1.x = brow[16];  b1.y = brow[DIM + 16];
    b2.x = brow[32];  b2.y = brow[DIM + 32];
    b3.x = brow[48];  b3.y = brow[DIM + 48];
    c0 = __builtin_amdgcn_wmma_f32_16x16x4_f32(false, a, false, b0, (short)0, c0, false, false);
    c1 = __builtin_amdgcn_wmma_f32_16x16x4_f32(false, a, false, b1, (short)0, c1, false, false);
    c2 = __builtin_amdgcn_wmma_f32_16x16x4_f32(false, a, false, b2, (short)0, c2, false, false);
    c3 = __builtin_amdgcn_wmma_f32_16x16x4_f32(false, a, false, b3, (short)0, c3, false, false);
  }

  float* cbase = C + (size_t)(tm * 16 + 8 * half) * DIM + ng * 64 + lr;
#pragma unroll
  for (int v = 0; v < 8; ++v) {
    cbase[(size_t)v * DIM +  0] = c0[v];
    cbase[(size_t)v * DIM + 16] = c1[v];
    cbase[(size_t)v * DIM + 32] = c2[v];
    cbase[(size_t)v * DIM + 48] = c3[v];
  }
}

// ---------------- GraphNorm (deterministic two-pass) ----------------
// pass 1: per-block partial sum & sum-of-squares per feature
__global__ void k_stats_part(const float* __restrict__ x, float* __restrict__ psum,
                             float* __restrict__ psq, int rows) {
  int d  = threadIdx.x;
  int b  = blockIdx.x;
  int r0 = b * STAT_ROWS;
  int r1 = min(r0 + STAT_ROWS, rows);
  float s = 0.0f, q = 0.0f;
  for (int r = r0; r < r1; ++r) {
    float v = x[(size_t)r * DIM + d];
    s += v;  q += v * v;
  }
  psum[(size_t)b * DIM + d] = s;
  psq [(size_t)b * DIM + d] = q;
}

// pass 2: reduce partials; derive affine s,t so norm = s*x + t
// out = g*(x - a*m)*rsqrt(var+eps) + b, var = E[x^2] - (2a - a^2)*m^2
__global__ void k_stats_reduce(const float* __restrict__ psum, const float* __restrict__ psq,
                               int nblk, const float* __restrict__ ga,
                               const float* __restrict__ gg, const float* __restrict__ gb,
                               float* __restrict__ sc, float* __restrict__ sh) {
  int d = threadIdx.x;
  float s = 0.0f, q = 0.0f;
  for (int b = 0; b < nblk; ++b) {
    s += psum[(size_t)b * DIM + d];
    q += psq [(size_t)b * DIM + d];
  }
  float m   = s * (1.0f / N_NODES);
  float ex2 = q * (1.0f / N_NODES);
  float a   = ga[d];
  float var = ex2 - (2.0f * a - a * a) * m * m;
  float scl = gg[d] * rsqrtf(var + EPS_GN);
  sc[d] = scl;
  sh[d] = gb[d] - scl * a * m;
}

// fused affine + activation: mode 0: leaky_relu(0.1)(s*x+t); mode 1: relu(res + s*x+t)
__global__ void k_affine_act(const float* __restrict__ x, float* __restrict__ h,
                             const float* __restrict__ sc, const float* __restrict__ sh,
                             const float* __restrict__ residual, int mode) {
  int i = blockIdx.x * blockDim.x + threadIdx.x;
  if (i >= N_NODES * DIM) return;
  int d = i & 255;
  float z = sc[d] * x[i] + sh[d];
  if (mode == 0) {
    h[i] = (z > 0.0f) ? z : 0.1f * z;
  } else {
    z += residual[i];
    h[i] = (z > 0.0f) ? z : 0.0f;
  }
}

// reduce per-block partial column sums -> single vector
__global__ void k_reduce_sum(const float* __restrict__ psum, int nblk, float* __restrict__ out) {
  int d = threadIdx.x;
  float s = 0.0f;
  for (int b = 0; b < nblk; ++b) s += psum[(size_t)b * DIM + d];
  out[d] = s;
}

// out[o] = softplus( mean_pool . lin_W[:,o] + lin_b[o] )
__global__ void k_final(const float* __restrict__ pooledsum, const float* __restrict__ W,
                        const float* __restrict__ bias, float* __restrict__ out) {
  int o = threadIdx.x;
  if (o >= OUTF) return;
  float acc = bias[o];
  for (int d = 0; d < DIM; ++d)
    acc += (pooledsum[d] * (1.0f / N_NODES)) * W[d * OUTF + o];
  out[o] = (acc > 20.0f) ? acc : log1pf(expf(acc));
}

// ---------------- host-side orchestration ----------------
extern "C" void kernel_launch(void* const* d_in, const int* in_sizes, int n_in,
                              void* d_out, int out_size, void* d_ws, size_t ws_size,
                              hipStream_t stream) {
  (void)in_sizes; (void)n_in; (void)out_size; (void)ws_size;

  // JAX pytree (sorted-key) flattening:
  // 0: edge_index (2,E) int32   1: feat (N,D) f32
  // per layer l: 2+5l:W  3+5l:b  4+5l:gn_a  5+5l:gn_b  6+5l:gn_g
  // 22: lin_W  23: lin_b
  const int*   edge = (const int*)d_in[0];
  const int*   erow = edge;
  const int*   ecol = edge + N_EDGES;
  const float* feat = (const float*)d_in[1];
  const float* linW = (const float*)d_in[2 + 5 * NLAYERS];
  const float* linB = (const float*)d_in[3 + 5 * NLAYERS];
  float* out = (float*)d_out;

  // workspace carve-up
  char* wsb = (char*)d_ws;
  int*   degi    = (int*)wsb;                       wsb += sizeof(int)   * N_NODES;
  int*   rowptr  = (int*)wsb;                       wsb += sizeof(int)   * (N_NODES + 1);
  int*   cursor  = (int*)wsb;                       wsb += sizeof(int)   * N_NODES;
  int*   csr_col = (int*)wsb;                       wsb += sizeof(int)   * N_EDGES;
  // keep float region 16B aligned
  wsb = (char*)(((size_t)wsb + 15) & ~(size_t)15);
  float* csr_w   = (float*)wsb;                     wsb += sizeof(float) * N_EDGES;
  float* dis     = (float*)wsb;                     wsb += sizeof(float) * N_NODES;
  float* wed     = (float*)wsb;                     wsb += sizeof(float) * N_EDGES;
  float* TX      = (float*)wsb;                     wsb += sizeof(float) * (size_t)N_NODES * KDIM;
  float* CONV    = (float*)wsb;                     wsb += sizeof(float) * (size_t)N_NODES * DIM;
  float* H       = (float*)wsb;                     wsb += sizeof(float) * (size_t)N_NODES * DIM;
  float* psum    = (float*)wsb;                     wsb += sizeof(float) * STAT_BLOCKS * DIM;
  float* psq     = (float*)wsb;                     wsb += sizeof(float) * STAT_BLOCKS * DIM;
  float* scv     = (float*)wsb;                     wsb += sizeof(float) * DIM;
  float* shv     = (float*)wsb;                     wsb += sizeof(float) * DIM;
  float* pooled  = (float*)wsb;                     wsb += sizeof(float) * DIM;

  const int TB = 256;
  const int gN    = (N_NODES + TB - 1) / TB;
  const int gE    = (N_EDGES + TB - 1) / TB;
  const int gND   = (N_NODES * DIM) / TB;           // 20000
  const int gProp = (N_NODES * 32) / TB;            // 2500 (wave per node)
  const dim3 gGemm(N_NODES / 32, 1);                // 625 blocks, 8 waves each

  // --- CSR build + edge weights (once per call) ---
  k_zero_i32<<<gN, TB, 0, stream>>>(degi, N_NODES);
  k_deg<<<gE, TB, 0, stream>>>(erow, degi);
  k_dis<<<gN, TB, 0, stream>>>(degi, dis);
  k_wedge<<<gE, TB, 0, stream>>>(erow, ecol, dis, wed);
  k_scan<<<1, TB, 0, stream>>>(degi, rowptr);
  k_copy_i32<<<gN, TB, 0, stream>>>(cursor, rowptr, N_NODES);
  k_csr_fill<<<gE, TB, 0, stream>>>(erow, ecol, wed, cursor, csr_col, csr_w);

  for (int l = 0; l < NLAYERS; ++l) {
    const float* W   = (const float*)d_in[2 + 5 * l];  // (K,D,D) == Wcat (KDIM x DIM)
    const float* b   = (const float*)d_in[3 + 5 * l];
    const float* gna = (const float*)d_in[4 + 5 * l];
    const float* gnb = (const float*)d_in[5 + 5 * l];
    const float* gng = (const float*)d_in[6 + 5 * l];
    const float* x = (l == 0) ? feat : H;

    // Chebyshev basis into TX blocks (fused recurrence, gather SpMM)
    k_copy<<<gND, TB, 0, stream>>>(TX + 0 * DIM, KDIM, x, DIM);
    k_prop<<<gProp, TB, 0, stream>>>(TX + 1 * DIM, TX + 0 * DIM, nullptr,
                                     rowptr, csr_col, csr_w, 1.0f, 0.0f);
    k_prop<<<gProp, TB, 0, stream>>>(TX + 2 * DIM, TX + 1 * DIM, TX + 0 * DIM,
                                     rowptr, csr_col, csr_w, 2.0f, -1.0f);
    k_prop<<<gProp, TB, 0, stream>>>(TX + 3 * DIM, TX + 2 * DIM, TX + 1 * DIM,
                                     rowptr, csr_col, csr_w, 2.0f, -1.0f);

    // CONV = TX @ Wcat + b   (fp32 WMMA, 16x64 strip per wave)
    k_gemm_wmma<<<gGemm, TB, 0, stream>>>(TX, W, b, CONV);

    // GraphNorm -> affine + activation
    k_stats_part<<<STAT_BLOCKS, TB, 0, stream>>>(CONV, psum, psq, N_NODES);
    k_stats_reduce<<<1, TB, 0, stream>>>(psum, psq, STAT_BLOCKS, gna, gng, gnb, scv, shv);
    if (l < NLAYERS - 1)
      k_affine_act<<<gND, TB, 0, stream>>>(CONV, H, scv, shv, nullptr, 0);
    else
      k_affine_act<<<gND, TB, 0, stream>>>(CONV, H, scv, shv, feat, 1);
  }

  // mean pool + linear + softplus
  k_stats_part<<<STAT_BLOCKS, TB, 0, stream>>>(H, psum, psq, N_NODES);
  k_reduce_sum<<<1, TB, 0, stream>>>(psum, STAT_BLOCKS, pooled);
  k_final<<<1, 32, 0, stream>>>(pooled, linW, linB, out);
}